// Net_71356586656067
// MI455X (gfx1250) — compile-verified
//
#include <hip/hip_runtime.h>

// ---------------------------------------------------------------------------
// MI455X (gfx1250) fused equivariant GNN edge update.
// All dense contractions -> V_WMMA_F32_16X16X4_F32 (full fp32 matrix pipe).
// Weights staged in the 320KB WGP LDS; per-wave 16-edge A tiles in LDS with
// bank-conflict-free padded strides (A: K+4, W: N+8 -> half-wave disjoint).
// ---------------------------------------------------------------------------

#define EE      320000          // edges
#define NTILES  (EE / 16)       // 20000 tiles of 16 edges
#define TBLK    2500            // NTILES / 8 waves per block
#define INV_SQ3 0.57735026918962576f
#define INV_SQ2 0.70710678118654752f

typedef float v2f __attribute__((ext_vector_type(2)));
typedef float v8f __attribute__((ext_vector_type(8)));

__device__ __forceinline__ v8f wmma4(v2f a, v2f b, v8f c) {
  // 8-arg pattern: (neg_a, A, neg_b, B, c_mod, C, reuse_a, reuse_b)
  return __builtin_amdgcn_wmma_f32_16x16x4_f32(false, a, false, b, (short)0, c,
                                               false, false);
}

// D(16x16) = A(16xK) * W(KxN)[:, ntile*16 .. +16), A/W in LDS.
// A frag (ISA 7.12.2): lane L: m=L&15, h=L>>4 -> a=(A[m][4kb+2h], A[m][4kb+2h+1])
// B frag (mirrored):   lane L: n=L&15, h=L>>4 -> b=(W[4kb+2h][n],  W[4kb+2h+1][n])
// D frag: VGPR r -> row (r + 8h), col n.
__device__ __forceinline__ v8f gemm16(const float* A, int sA, const float* W,
                                      int sW, int K, int ntile, int lane) {
  const int m = lane & 15;
  const int h = lane >> 4;
  const float* ar = A + m * sA + 2 * h;
  const float* wc = W + ntile * 16 + m + 2 * h * sW;
  v8f acc = {0.f, 0.f, 0.f, 0.f, 0.f, 0.f, 0.f, 0.f};
#pragma unroll 4
  for (int kb = 0; kb < (K >> 2); ++kb) {
    v2f a = *(const v2f*)(ar + 4 * kb);
    v2f b;
    b.x = wc[4 * kb * sW];
    b.y = wc[4 * kb * sW + sW];
    acc = wmma4(a, b, acc);
  }
  return acc;
}

__device__ __forceinline__ float sigm(float x) { return 1.f / (1.f + __expf(-x)); }

// ============================ Kernel A1 =====================================
// es = edge_s@Wpre_s + b ; s_in = [node_s[i] | node_s[j] | es]
// zs = s_in@Wtp_ss (E x 96), zsv = s_in@Wtp_sv (E x 32)
__global__ void __launch_bounds__(256) k_a1(
    const float* __restrict__ node_s, const float* __restrict__ edge_s,
    const int* __restrict__ eidx, const float* __restrict__ Wpre_s,
    const float* __restrict__ bpre_s, const float* __restrict__ Wtp_ss,
    const float* __restrict__ Wtp_sv, float* __restrict__ zs,
    float* __restrict__ zsv) {
  extern __shared__ float sm[];
  float* Wps = sm;                 // 64 x 72
  float* Wss = Wps + 64 * 72;      // 192 x 104
  float* Wsv = Wss + 192 * 104;    // 192 x 40
  float* stage = Wsv + 192 * 40;   // 8 waves x 16 x 196
  const int tid = threadIdx.x, nth = blockDim.x;
  for (int i = tid; i < 64 * 64; i += nth) Wps[(i >> 6) * 72 + (i & 63)] = Wpre_s[i];
  for (int i = tid; i < 192 * 96; i += nth) { int k = i / 96, n = i - k * 96; Wss[k * 104 + n] = Wtp_ss[i]; }
  for (int i = tid; i < 192 * 32; i += nth) { int k = i >> 5, n = i & 31; Wsv[k * 40 + n] = Wtp_sv[i]; }
  __syncthreads();
  const int lane = tid & 31, widx = tid >> 5;
  const int m = lane & 15, h = lane >> 4;
  float* sb = stage + widx * (16 * 196);
  const int gw = blockIdx.x * 8 + widx, nw = gridDim.x * 8;
  for (int tile = gw; tile < NTILES; tile += nw) {
    const int base = tile * 16;
    // stage edge_s rows into cols 128..191
    for (int r = 0; r < 16; ++r) {
      const int e = base + r;
      sb[r * 196 + 128 + lane] = edge_s[e * 64 + lane];
      sb[r * 196 + 160 + lane] = edge_s[e * 64 + 32 + lane];
    }
    v8f dp[4];
    for (int t = 0; t < 4; ++t) dp[t] = gemm16(sb + 128, 196, Wps, 72, 64, t, lane);
    for (int t = 0; t < 4; ++t) {
      const int n = m + t * 16;
      const float bb = bpre_s[n];
#pragma unroll
      for (int r = 0; r < 8; ++r) sb[(r + 8 * h) * 196 + 128 + n] = dp[t][r] + bb;
    }
    // gather node rows into cols 0..127
    for (int r = 0; r < 16; ++r) {
      const int e = base + r;
      const int ii = eidx[e], jj = eidx[EE + e];
      sb[r * 196 + lane]      = node_s[ii * 64 + lane];
      sb[r * 196 + 32 + lane] = node_s[ii * 64 + 32 + lane];
      sb[r * 196 + 64 + lane] = node_s[jj * 64 + lane];
      sb[r * 196 + 96 + lane] = node_s[jj * 64 + 32 + lane];
    }
    for (int t = 0; t < 6; ++t) {
      v8f d = gemm16(sb, 196, Wss, 104, 192, t, lane);
      const int n = m + t * 16;
#pragma unroll
      for (int r = 0; r < 8; ++r) zs[(base + r + 8 * h) * 96 + n] = d[r];
    }
    for (int t = 0; t < 2; ++t) {
      v8f d = gemm16(sb, 196, Wsv, 40, 192, t, lane);
      const int n = m + t * 16;
#pragma unroll
      for (int r = 0; r < 8; ++r) zsv[(base + r + 8 * h) * 32 + n] = d[r];
    }
  }
}

// ============================ Kernel FC =====================================
// w = fc3(silu(fc2(silu(fc1(emb)))))  -> [E,96]
__global__ void __launch_bounds__(256) k_fc(
    const float* __restrict__ emb, const float* __restrict__ fc1,
    const float* __restrict__ fcb1, const float* __restrict__ fc2,
    const float* __restrict__ fcb2, const float* __restrict__ fc3,
    const float* __restrict__ fcb3, float* __restrict__ wout) {
  extern __shared__ float sm[];
  float* W1 = sm;                // 64 x 72
  float* W2 = W1 + 64 * 72;      // 64 x 72
  float* W3 = W2 + 64 * 72;      // 64 x 104
  float* stage = W3 + 64 * 104;  // 8 waves x 2 x 16 x 68
  const int tid = threadIdx.x, nth = blockDim.x;
  for (int i = tid; i < 64 * 64; i += nth) {
    int k = i >> 6, n = i & 63;
    W1[k * 72 + n] = fc1[i];
    W2[k * 72 + n] = fc2[i];
  }
  for (int i = tid; i < 64 * 96; i += nth) { int k = i / 96, n = i - k * 96; W3[k * 104 + n] = fc3[i]; }
  __syncthreads();
  const int lane = tid & 31, widx = tid >> 5;
  const int m = lane & 15, h = lane >> 4;
  float* bA = stage + widx * (2 * 16 * 68);
  float* bB = bA + 16 * 68;
  const int gw = blockIdx.x * 8 + widx, nw = gridDim.x * 8;
  for (int tile = gw; tile < NTILES; tile += nw) {
    const int base = tile * 16;
    for (int r = 0; r < 16; ++r) {
      const int e = base + r;
      bA[r * 68 + lane]      = emb[e * 64 + lane];
      bA[r * 68 + 32 + lane] = emb[e * 64 + 32 + lane];
    }
    v8f d[4];
    for (int t = 0; t < 4; ++t) d[t] = gemm16(bA, 68, W1, 72, 64, t, lane);
    for (int t = 0; t < 4; ++t) {
      const int n = m + t * 16; const float bb = fcb1[n];
#pragma unroll
      for (int r = 0; r < 8; ++r) { float x = d[t][r] + bb; bB[(r + 8 * h) * 68 + n] = x * sigm(x); }
    }
    for (int t = 0; t < 4; ++t) d[t] = gemm16(bB, 68, W2, 72, 64, t, lane);
    for (int t = 0; t < 4; ++t) {
      const int n = m + t * 16; const float bb = fcb2[n];
#pragma unroll
      for (int r = 0; r < 8; ++r) { float x = d[t][r] + bb; bA[(r + 8 * h) * 68 + n] = x * sigm(x); }
    }
    for (int t = 0; t < 6; ++t) {
      v8f dd = gemm16(bA, 68, W3, 104, 64, t, lane);
      const int n = m + t * 16; const float bb = fcb3[n];
#pragma unroll
      for (int r = 0; r < 8; ++r) wout[(base + r + 8 * h) * 96 + n] = dd[r] + bb;
    }
  }
}

// ============================ Kernel A2 =====================================
// Vector path + gating: produces g_s [E,64], g_v [E,96] (k*3+c layout)
__global__ void __launch_bounds__(256) k_a2(
    const float* __restrict__ node_v, const float* __restrict__ edge_v,
    const int* __restrict__ eidx, const float* __restrict__ sh0,
    const float* __restrict__ sh1, const float* __restrict__ Wpre_v,
    const float* __restrict__ Wtp_vv, const float* __restrict__ Wtp_vs,
    const float* __restrict__ zs_in, const float* __restrict__ zsv_in,
    const float* __restrict__ wfc, float* __restrict__ gs,
    float* __restrict__ gv) {
  extern __shared__ float sm[];
  float* Wpv = sm;               // 32 x 40
  float* Wvv = Wpv + 32 * 40;    // 96 x 104
  float* Wvs = Wvv + 96 * 104;   // 96 x 40
  float* stage = Wvs + 96 * 40;  // 8 waves x 6528
  const int tid = threadIdx.x, nth = blockDim.x;
  for (int i = tid; i < 32 * 32; i += nth) Wpv[(i >> 5) * 40 + (i & 31)] = Wpre_v[i];
  for (int i = tid; i < 96 * 96; i += nth) { int k = i / 96, n = i - k * 96; Wvv[k * 104 + n] = Wtp_vv[i]; }
  for (int i = tid; i < 96 * 32; i += nth) { int k = i >> 5, n = i & 31; Wvs[k * 40 + n] = Wtp_vs[i]; }
  __syncthreads();
  const int lane = tid & 31, widx = tid >> 5;
  const int m = lane & 15, h = lane >> 4;
  float* vb = stage + widx * 6528;  // 16 x 100 current component v_in
  float* vd = vb + 1600;            // 16 x 100 vdot accumulator
  float* zvb = vd + 1600;           // 16 x 104 (3 x 32) tp_vs parts
  float* zsb = zvb + 16 * 104;      // 16 x 104 z_s
  const int gw = blockIdx.x * 8 + widx, nw = gridDim.x * 8;
  for (int tile = gw; tile < NTILES; tile += nw) {
    const int base = tile * 16;
    for (int idx = lane; idx < 1600; idx += 32) vd[idx] = 0.f;
    for (int c = 0; c < 3; ++c) {
      for (int r = 0; r < 16; ++r) {
        const int e = base + r;
        vb[r * 100 + 64 + lane] = edge_v[e * 96 + lane * 3 + c];
      }
      v8f dv[2];
      for (int t = 0; t < 2; ++t) dv[t] = gemm16(vb + 64, 100, Wpv, 40, 32, t, lane);
      for (int t = 0; t < 2; ++t) {
        const int n = m + t * 16;
#pragma unroll
        for (int r = 0; r < 8; ++r) vb[(r + 8 * h) * 100 + 64 + n] = dv[t][r];
      }
      for (int r = 0; r < 16; ++r) {
        const int e = base + r;
        const int ii = eidx[e], jj = eidx[EE + e];
        vb[r * 100 + lane]      = node_v[ii * 96 + lane * 3 + c];
        vb[r * 100 + 32 + lane] = node_v[jj * 96 + lane * 3 + c];
      }
      for (int idx = lane; idx < 16 * 96; idx += 32) {
        const int r = idx / 96, k = idx - r * 96;
        vd[r * 100 + k] += vb[r * 100 + k] * sh1[(base + r) * 3 + c];
      }
      for (int t = 0; t < 2; ++t) {
        v8f d = gemm16(vb, 100, Wvs, 40, 96, t, lane);
        const int n = m + t * 16;
#pragma unroll
        for (int r = 0; r < 8; ++r) zvb[(r + 8 * h) * 104 + c * 32 + n] = d[r];
      }
    }
    for (int t = 0; t < 6; ++t) {
      v8f d = gemm16(vd, 100, Wvv, 104, 96, t, lane);
      const int n = m + t * 16;
#pragma unroll
      for (int r = 0; r < 8; ++r) {
        const int e = base + r + 8 * h;
        zsb[(r + 8 * h) * 104 + n] =
            (zs_in[e * 96 + n] * sh0[e] + d[r] * INV_SQ3) * INV_SQ2;
      }
    }
    for (int idx = lane; idx < 16 * 64; idx += 32) {
      const int r = idx >> 6, k = idx & 63, e = base + r;
      const float z = zsb[r * 104 + k];
      gs[e * 64 + k] = z * sigm(z) * wfc[e * 96 + k];
    }
    for (int idx = lane; idx < 16 * 32; idx += 32) {
      const int r = idx >> 5, k = idx & 31, e = base + r;
      const float gate = sigm(zsb[r * 104 + 64 + k]);
      const float wk = wfc[e * 96 + 64 + k];
      const float zse = zsv_in[e * 32 + k];
      const float s0 = sh0[e];
      for (int c = 0; c < 3; ++c) {
        const float zv =
            (zse * sh1[e * 3 + c] + zvb[r * 104 + c * 32 + k] * s0) * INV_SQ2;
        gv[e * 96 + k * 3 + c] = zv * gate * wk;
      }
    }
  }
}

// ============================ Kernel B ======================================
// o = lin_post(g) + self_connection(edge, one_hot)
__global__ void __launch_bounds__(256) k_b(
    const float* __restrict__ edge_s, const float* __restrict__ edge_v,
    const float* __restrict__ onehot, const float* __restrict__ gs,
    const float* __restrict__ gv, const float* __restrict__ Wpost_s,
    const float* __restrict__ bpost_s, const float* __restrict__ Wpost_v,
    const float* __restrict__ Wsc_s, const float* __restrict__ Wsc_v,
    float* __restrict__ os, float* __restrict__ ov) {
  extern __shared__ float sm[];
  float* Wps = sm;                // 64 x 72
  float* Wpv = Wps + 64 * 72;     // 32 x 40
  float* Wss = Wpv + 32 * 40;     // 64 x 264  (Wsc_s as [64][q*64+k])
  float* Wsv = Wss + 64 * 264;    // 32 x 136  (Wsc_v as [32][q*32+k])
  float* stage = Wsv + 32 * 136;  // 8 waves x 1728
  const int tid = threadIdx.x, nth = blockDim.x;
  for (int i = tid; i < 64 * 64; i += nth) Wps[(i >> 6) * 72 + (i & 63)] = Wpost_s[i];
  for (int i = tid; i < 32 * 32; i += nth) Wpv[(i >> 5) * 40 + (i & 31)] = Wpost_v[i];
  for (int i = tid; i < 64 * 256; i += nth) { int k = i >> 8, n = i & 255; Wss[k * 264 + n] = Wsc_s[i]; }
  for (int i = tid; i < 32 * 128; i += nth) { int k = i >> 7, n = i & 127; Wsv[k * 136 + n] = Wsc_v[i]; }
  __syncthreads();
  const int lane = tid & 31, widx = tid >> 5;
  const int m = lane & 15, h = lane >> 4;
  float* ab = stage + widx * 1728;  // 16 x 68
  float* vs = ab + 16 * 68;         // 16 x 36
  float* oh = vs + 16 * 36;         // 16 x 4
  const int gw = blockIdx.x * 8 + widx, nw = gridDim.x * 8;
  for (int tile = gw; tile < NTILES; tile += nw) {
    const int base = tile * 16;
    for (int idx = lane; idx < 64; idx += 32) oh[idx] = onehot[base * 4 + idx];
    // ---- scalar channel ----
    for (int r = 0; r < 16; ++r) {
      const int e = base + r;
      ab[r * 68 + lane]      = gs[e * 64 + lane];
      ab[r * 68 + 32 + lane] = gs[e * 64 + 32 + lane];
    }
    v8f ps[4];
    for (int t = 0; t < 4; ++t) ps[t] = gemm16(ab, 68, Wps, 72, 64, t, lane);
    for (int r = 0; r < 16; ++r) {
      const int e = base + r;
      ab[r * 68 + lane]      = edge_s[e * 64 + lane];
      ab[r * 68 + 32 + lane] = edge_s[e * 64 + 32 + lane];
    }
    for (int t = 0; t < 4; ++t)
      for (int q = 0; q < 4; ++q) {
        v8f d = gemm16(ab, 68, Wss, 264, 64, q * 4 + t, lane);
#pragma unroll
        for (int r = 0; r < 8; ++r) ps[t][r] += oh[(r + 8 * h) * 4 + q] * d[r];
      }
    for (int t = 0; t < 4; ++t) {
      const int n = m + t * 16; const float bb = bpost_s[n];
#pragma unroll
      for (int r = 0; r < 8; ++r) os[(base + r + 8 * h) * 64 + n] = ps[t][r] + bb;
    }
    // ---- vector channel, per component ----
    for (int c = 0; c < 3; ++c) {
      for (int r = 0; r < 16; ++r) vs[r * 36 + lane] = gv[(base + r) * 96 + lane * 3 + c];
      v8f pv[2];
      for (int t = 0; t < 2; ++t) pv[t] = gemm16(vs, 36, Wpv, 40, 32, t, lane);
      for (int r = 0; r < 16; ++r) vs[r * 36 + lane] = edge_v[(base + r) * 96 + lane * 3 + c];
      for (int t = 0; t < 2; ++t)
        for (int q = 0; q < 4; ++q) {
          v8f d = gemm16(vs, 36, Wsv, 136, 32, q * 2 + t, lane);
#pragma unroll
          for (int r = 0; r < 8; ++r) pv[t][r] += oh[(r + 8 * h) * 4 + q] * d[r];
        }
      for (int t = 0; t < 2; ++t) {
        const int n = m + t * 16;
#pragma unroll
        for (int r = 0; r < 8; ++r) ov[(base + r + 8 * h) * 96 + n * 3 + c] = pv[t][r];
      }
    }
  }
}

// ============================ Stats kernels =================================
// stats layout (floats): cnt[8] | sum[8*64] | ss2[8] | sv2[8] | mean[8*64] | invs[8] | invv[8]
__global__ void __launch_bounds__(256) k_stats(const float* __restrict__ os,
                                               const float* __restrict__ ov,
                                               const int* __restrict__ eidx,
                                               const int* __restrict__ batch,
                                               float* __restrict__ stats) {
  __shared__ float lc[8], lsum[512], ls2[8], lv2[8];
  const int tid = threadIdx.x;
  for (int i = tid; i < 512; i += blockDim.x) lsum[i] = 0.f;
  if (tid < 8) { lc[tid] = 0.f; ls2[tid] = 0.f; lv2[tid] = 0.f; }
  __syncthreads();
  const int stride = gridDim.x * blockDim.x;
  for (int e = blockIdx.x * blockDim.x + tid; e < EE; e += stride) {
    const int g = batch[eidx[e]];
    atomicAdd(&lc[g], 1.f);
    float a2 = 0.f;
    for (int k = 0; k < 64; ++k) {
      const float v = os[e * 64 + k];
      atomicAdd(&lsum[g * 64 + k], v);
      a2 += v * v;
    }
    atomicAdd(&ls2[g], a2);
    float b2 = 0.f;
    for (int k = 0; k < 96; ++k) { const float v = ov[e * 96 + k]; b2 += v * v; }
    atomicAdd(&lv2[g], b2);
  }
  __syncthreads();
  for (int i = tid; i < 512; i += blockDim.x) atomicAdd(&stats[8 + i], lsum[i]);
  if (tid < 8) {
    atomicAdd(&stats[tid], lc[tid]);
    atomicAdd(&stats[520 + tid], ls2[tid]);
    atomicAdd(&stats[528 + tid], lv2[tid]);
  }
}

__global__ void k_fin(float* __restrict__ stats) {
  const int t = threadIdx.x;
  if (t < 512) {
    const int g = t >> 6;
    const float c = fmaxf(stats[g], 1.f);
    stats[536 + t] = stats[8 + t] / c;
  }
  __syncthreads();
  if (t < 8) {
    const float c = fmaxf(stats[t], 1.f);
    float sq = 0.f;
    for (int k = 0; k < 64; ++k) { const float mu = stats[536 + t * 64 + k]; sq += mu * mu; }
    const float var_s = stats[520 + t] / (c * 64.f) - sq / 64.f;
    stats[1048 + t] = rsqrtf(var_s + 1e-5f);
    const float var_v = stats[528 + t] / (c * 32.f);
    stats[1056 + t] = rsqrtf(var_v + 1e-5f);
  }
}

__global__ void __launch_bounds__(256) k_apply(
    const float* __restrict__ os, const float* __restrict__ ov,
    const float* __restrict__ edge_s, const float* __restrict__ edge_v,
    const int* __restrict__ eidx, const int* __restrict__ batch,
    const float* __restrict__ stats, const float* __restrict__ ln_g_s,
    const float* __restrict__ ln_b_s, const float* __restrict__ ln_g_v,
    float* __restrict__ out) {
  const long long id = (long long)blockIdx.x * blockDim.x + threadIdx.x;
  const long long total = (long long)EE * 160;
  if (id >= total) return;
  const int e = (int)(id / 160);
  const int r = (int)(id - (long long)e * 160);
  const int g = batch[eidx[e]];
  float val;
  if (r < 64) {
    val = (os[e * 64 + r] - stats[536 + g * 64 + r]) * stats[1048 + g] * ln_g_s[r]
          + ln_b_s[r] + edge_s[e * 64 + r];
  } else {
    const int rr = r - 64;
    const int k = rr / 3;
    val = ov[e * 96 + rr] * stats[1056 + g] * ln_g_v[k] + edge_v[e * 96 + rr];
  }
  out[id] = val;
}

// ============================ Launcher ======================================
extern "C" void kernel_launch(void* const* d_in, const int* in_sizes, int n_in,
                              void* d_out, int out_size, void* d_ws,
                              size_t ws_size, hipStream_t stream) {
  const float* node_s   = (const float*)d_in[0];
  const float* node_v   = (const float*)d_in[1];
  const float* edge_s   = (const float*)d_in[2];
  const float* edge_v   = (const float*)d_in[3];
  const float* edge_sh0 = (const float*)d_in[4];
  const float* edge_sh1 = (const float*)d_in[5];
  const float* onehot   = (const float*)d_in[6];
  const float* emb      = (const float*)d_in[7];
  const int*   eidx     = (const int*)d_in[8];
  const int*   batch    = (const int*)d_in[9];
  const float* Wsc_s  = (const float*)d_in[10];
  const float* Wsc_v  = (const float*)d_in[11];
  const float* Wpre_s = (const float*)d_in[12];
  const float* bpre_s = (const float*)d_in[13];
  const float* Wpre_v = (const float*)d_in[14];
  const float* Wtp_ss = (const float*)d_in[15];
  const float* Wtp_vv = (const float*)d_in[16];
  const float* Wtp_sv = (const float*)d_in[17];
  const float* Wtp_vs = (const float*)d_in[18];
  const float* fc1 = (const float*)d_in[19];
  const float* fcb1 = (const float*)d_in[20];
  const float* fc2 = (const float*)d_in[21];
  const float* fcb2 = (const float*)d_in[22];
  const float* fc3 = (const float*)d_in[23];
  const float* fcb3 = (const float*)d_in[24];
  const float* Wpost_s = (const float*)d_in[25];
  const float* bpost_s = (const float*)d_in[26];
  const float* Wpost_v = (const float*)d_in[27];
  const float* ln_g_s = (const float*)d_in[28];
  const float* ln_b_s = (const float*)d_in[29];
  const float* ln_g_v = (const float*)d_in[30];

  // Workspace layout (floats). Lifetimes overlap: region1 (zs|zsv|wfc, 224/edge)
  // is dead after k_a2 and reused for o_s|o_v (160/edge); region2 (g, 160/edge)
  // is dead after k_b and reused for the 1064-float stats block.
  float* ws  = (float*)d_ws;
  const size_t E = EE;
  float* zs  = ws;                // E*96
  float* zsv = zs + E * 96;       // E*32
  float* wfc = zsv + E * 32;      // E*96
  float* gs  = wfc + E * 96;      // E*64
  float* gv  = gs + E * 64;       // E*96
  float* os  = ws;                // reuse region1
  float* ov  = os + E * 64;
  float* stats = gs;              // reuse region2 (1064 floats)

  constexpr int A1_LDS = (64 * 72 + 192 * 104 + 192 * 40 + 8 * 16 * 196) * 4; // 229376
  constexpr int FC_LDS = (64 * 72 * 2 + 64 * 104 + 8 * 2 * 16 * 68) * 4;      // 133120
  constexpr int A2_LDS = (32 * 40 + 96 * 104 + 96 * 40 + 8 * 6528) * 4;       // 269312
  constexpr int B_LDS  = (64 * 72 + 32 * 40 + 64 * 264 + 32 * 136 + 8 * 1728) * 4; // 163840

  (void)hipFuncSetAttribute((const void*)k_a1, hipFuncAttributeMaxDynamicSharedMemorySize, A1_LDS);
  (void)hipFuncSetAttribute((const void*)k_fc, hipFuncAttributeMaxDynamicSharedMemorySize, FC_LDS);
  (void)hipFuncSetAttribute((const void*)k_a2, hipFuncAttributeMaxDynamicSharedMemorySize, A2_LDS);
  (void)hipFuncSetAttribute((const void*)k_b,  hipFuncAttributeMaxDynamicSharedMemorySize, B_LDS);

  k_a1<<<TBLK, 256, A1_LDS, stream>>>(node_s, edge_s, eidx, Wpre_s, bpre_s,
                                      Wtp_ss, Wtp_sv, zs, zsv);
  k_fc<<<TBLK, 256, FC_LDS, stream>>>(emb, fc1, fcb1, fc2, fcb2, fc3, fcb3, wfc);
  k_a2<<<TBLK, 256, A2_LDS, stream>>>(node_v, edge_v, eidx, edge_sh0, edge_sh1,
                                      Wpre_v, Wtp_vv, Wtp_vs, zs, zsv, wfc, gs, gv);
  k_b<<<TBLK, 256, B_LDS, stream>>>(edge_s, edge_v, onehot, gs, gv, Wpost_s,
                                    bpost_s, Wpost_v, Wsc_s, Wsc_v, os, ov);
  (void)hipMemsetAsync(stats, 0, 536 * sizeof(float), stream);
  k_stats<<<1024, 256, 0, stream>>>(os, ov, eidx, batch, stats);
  k_fin<<<1, 512, 0, stream>>>(stats);
  {
    const long long total = (long long)EE * 160;
    const int blocks = (int)((total + 255) / 256);
    k_apply<<<blocks, 256, 0, stream>>>(os, ov, edge_s, edge_v, eidx, batch,
                                        stats, ln_g_s, ln_b_s, ln_g_v,
                                        (float*)d_out);
  }
}